// Upscale_15358803050749
// MI455X (gfx1250) — compile-verified
//
#include <hip/hip_runtime.h>

// upfirdn2d up=2, 4x4 kernel (StyleGAN2 upsample), fp32.
// Memory-bound: 64 MiB read + 256 MiB written ~= 336 MB @ 23.3 TB/s => ~14.4us
// floor; 0.54 GFLOP of FMA is negligible => no WMMA (fp32, 4-tap stencil).
// CDNA5 paths: async global->LDS staging (b128 + b32, ASYNCcnt,
// s_wait_asynccnt) and non-temporal B128 streaming stores for the output.

#define TW 32                  // input tile width  (cells)
#define TH 32                  // input tile height (cells)
#define LDS_W 40               // row stride (floats): 160B, 16B-aligned rows
#define LDS_H (TH + 2)         // 34 rows (1-halo top/bottom)
#define COL0 4                 // LDS col of input col gx0 (16B aligned)
// left halo at col 3, data cols 4..35, right halo at col 36

typedef __attribute__((ext_vector_type(4))) float floatx4;

__global__ __launch_bounds__(256)
void Upscale_15358803050749_kernel(const float* __restrict__ x,
                                   const float* __restrict__ kern,
                                   float* __restrict__ out,
                                   int H, int W) {
    __shared__ __align__(16) float tile[LDS_H * LDS_W];

    const int tid   = threadIdx.x;
    const int gx0   = blockIdx.x * TW;    // input-tile origin (col)
    const int gy0   = blockIdx.y * TH;    // input-tile origin (row)
    const int plane = blockIdx.z;         // n*c plane index

    const float* __restrict__ xp = x + (size_t)plane * H * W;
    float* __restrict__ op = out + (size_t)plane * (size_t)(4 * H * W);
    const int OW = 2 * W;

    // ---- Stage (TH+2)x(TW+2) tile into LDS with async copies ----
    // Interior of each row: 8x B128 asyncs (16B-aligned global & LDS addrs).
    // Halo columns: B32 asyncs. Out-of-image: zero-fill via DS store
    // (disjoint addresses from async targets -> no ordering hazard).
    const uint32_t lds_base = (uint32_t)(uintptr_t)(&tile[0]);
    const int NB128 = LDS_H * (TW / 4);        // 34*8 = 272 vector copies
    const int NHALO = LDS_H * 2;               // 68 scalar halo copies
    for (int i = tid; i < NB128 + NHALO; i += 256) {
        if (i < NB128) {
            const int r = i >> 3, q = i & 7;   // row, quad-within-row
            const int gy = gy0 + r - 1;
            const int le = r * LDS_W + COL0 + 4 * q;
            if ((unsigned)gy < (unsigned)H) {
                const uint64_t ga =
                    (uint64_t)(uintptr_t)(xp + (size_t)gy * W + gx0 + 4 * q);
                const uint32_t la = lds_base + (uint32_t)le * 4u;
                asm volatile("global_load_async_to_lds_b128 %0, %1, off"
                             :: "v"(la), "v"(ga) : "memory");
            } else {
                floatx4 z = {0.f, 0.f, 0.f, 0.f};
                *(floatx4*)(&tile[le]) = z;
            }
        } else {
            const int h = i - NB128;
            const int r = h >> 1, side = h & 1;
            const int gy = gy0 + r - 1;
            const int gx = side ? (gx0 + TW) : (gx0 - 1);
            const int le = r * LDS_W + (side ? (COL0 + TW) : (COL0 - 1));
            if ((unsigned)gy < (unsigned)H && (unsigned)gx < (unsigned)W) {
                const uint64_t ga =
                    (uint64_t)(uintptr_t)(xp + (size_t)gy * W + gx);
                const uint32_t la = lds_base + (uint32_t)le * 4u;
                asm volatile("global_load_async_to_lds_b32 %0, %1, off"
                             :: "v"(la), "v"(ga) : "memory");
            } else {
                tile[le] = 0.0f;
            }
        }
    }

    // Kernel taps (uniform, 16 floats) — fetched while async copies fly.
    float K[16];
#pragma unroll
    for (int i = 0; i < 16; ++i) K[i] = kern[i];

    // Wave-local async completion, then block-wide visibility.
    asm volatile("s_wait_asynccnt 0x0" ::: "memory");
    __syncthreads();

    // ---- Compute: each thread owns a 2x2 cell block -> 4x4 outputs ----
    const int tx = tid & 15;          // cell-pair col: cells 2tx, 2tx+1
    const int ty = tid >> 4;          // cell-pair row: cells 2ty, 2ty+1
    const int lt = 2 * ty;            // first cell row in tile (0..30)

    // 4x4 input neighborhood: rows lt-1..lt+2 (LDS rows lt..lt+3),
    // cols 2tx-1..2tx+2 (LDS cols 3+2tx..6+2tx).
    float X[4][4];
#pragma unroll
    for (int a = 0; a < 4; ++a)
#pragma unroll
        for (int b = 0; b < 4; ++b)
            X[a][b] = tile[(lt + a) * LDS_W + (COL0 - 1 + 2 * tx + b)];

    const int s4 = 2 * (gx0 + 2 * tx);         // first output col (16B aligned)
#pragma unroll
    for (int v = 0; v < 2; ++v) {              // cell row within 2x2 block
#pragma unroll
        for (int w = 0; w < 2; ++w) {          // output-row phase
            floatx4 o;
#pragma unroll
            for (int u = 0; u < 2; ++u) {      // cell col within 2x2 block
                float e0, e1;
                if (w == 0) {   // even output row: rows {t:K[1], t-1:K[3]}
                    e0 = K[1*4+1]*X[1+v][1+u] + K[1*4+3]*X[1+v][u]
                       + K[3*4+1]*X[v][1+u]   + K[3*4+3]*X[v][u];
                    e1 = K[1*4+2]*X[1+v][1+u] + K[1*4+0]*X[1+v][2+u]
                       + K[3*4+2]*X[v][1+u]   + K[3*4+0]*X[v][2+u];
                } else {        // odd output row: rows {t:K[2], t+1:K[0]}
                    e0 = K[2*4+1]*X[1+v][1+u] + K[2*4+3]*X[1+v][u]
                       + K[0*4+1]*X[2+v][1+u] + K[0*4+3]*X[2+v][u];
                    e1 = K[2*4+2]*X[1+v][1+u] + K[2*4+0]*X[1+v][2+u]
                       + K[0*4+2]*X[2+v][1+u] + K[0*4+0]*X[2+v][2+u];
                }
                o[2*u]     = e0;
                o[2*u + 1] = e1;
            }
            const int orow = 2 * (gy0 + lt + v) + w;
            float* dst = op + (size_t)orow * OW + s4;
            // Write-once 256 MiB output: stream past L2 (th:TH_STORE_NT),
            // B128 per lane, contiguous 256B per 16-lane group.
            __builtin_nontemporal_store(o, (floatx4*)dst);
        }
    }
}

extern "C" void kernel_launch(void* const* d_in, const int* in_sizes, int n_in,
                              void* d_out, int out_size, void* d_ws, size_t ws_size,
                              hipStream_t stream) {
    const float* x = (const float*)d_in[0];   // (8,128,128,128) fp32
    const float* k = (const float*)d_in[1];   // (4,4) fp32
    float* out = (float*)d_out;               // (8,128,256,256) fp32

    const int H = 128, W = 128;
    const int nc = in_sizes[0] / (H * W);     // 1024 planes

    dim3 grid(W / TW, H / TH, nc);            // 4 x 4 x 1024 blocks
    dim3 block(256, 1, 1);                    // 8 wave32s
    Upscale_15358803050749_kernel<<<grid, block, 0, stream>>>(x, k, out, H, W);
}